// UpsamplerAttention_41918880808998
// MI455X (gfx1250) — compile-verified
//
#include <hip/hip_runtime.h>
#include <hip/hip_bf16.h>

#define NCTX 3
#define BB 8
#define CC 128
#define CF 256
#define HH 64
#define WW 64

typedef __attribute__((ext_vector_type(2))) float v2f;
typedef __attribute__((ext_vector_type(8))) float v8f;

__device__ __forceinline__ v8f wmma4(v2f a, v2f b, v8f c) {
  // D = A(16x4) * B(4x16) + C(16x16), full fp32
  return __builtin_amdgcn_wmma_f32_16x16x4_f32(false, a, false, b, (short)0, c, false, false);
}

// ---------------------------------------------------------------------------
// Kernel W: transpose Wv (Co,Ci,3,3) -> tap-major (9, Co, Ci) so conv A-operands
// are contiguous along Ci (enables b64 A loads in the projection kernel).
// ---------------------------------------------------------------------------
__global__ void wv_transpose_kernel(const float* __restrict__ wv,
                                    float* __restrict__ wvT) {
  int idx = blockIdx.x * 256 + threadIdx.x;
  if (idx >= CC * CF * 9) return;
  int o   = idx / (CF * 9);
  int rem = idx % (CF * 9);
  int ci  = rem / 9;
  int tap = rem % 9;
  wvT[((size_t)tap * CC + o) * CF + ci] = wv[idx];
}

// ---------------------------------------------------------------------------
// Kernel A: projections from mainstream.
//   q    = Wf * x + bf                (stored to ws)
//   vals = conv3x3(x, Wv) + bv       (stored to ws)
//   lmain= sum_c (Wk*x+bk)_c * q_c   (stored to ws)
// One WG = one (b, h) row, 32 pixels wide. 8 waves x (16 M x 32 N) tiles.
// Input tile (3 rows x 256 ci x 34 cols) staged in LDS, padded to 40 floats/row.
// ---------------------------------------------------------------------------
__global__ void __launch_bounds__(256)
proj_kernel(const float* __restrict__ x,
            const float* __restrict__ wf, const float* __restrict__ bf,
            const float* __restrict__ wk, const float* __restrict__ bk,
            const float* __restrict__ wvT, const float* __restrict__ bv,
            float* __restrict__ q_out, float* __restrict__ v_out,
            float* __restrict__ lmain_out) {
  const int w0 = blockIdx.x * 32;
  const int h  = blockIdx.y;
  const int b  = blockIdx.z;
  const int tid  = threadIdx.x;
  const int wave = tid >> 5, lane = tid & 31;
  const int lrow = lane & 15;
  const int khalf = (lane >> 4) * 2;

  __shared__ float xs[3][CF][40];
  __shared__ float lm_s[32];
  if (tid < 32) lm_s[tid] = 0.0f;

  // cooperative, coalesced global->LDS with zero padding at image borders
  for (int idx = tid; idx < 3 * CF * 34; idx += 256) {
    int row = idx / (CF * 34);
    int rem = idx % (CF * 34);
    int ci  = rem / 34;
    int wl  = rem % 34;
    int gh = h - 1 + row;
    int gw = w0 - 1 + wl;
    float v = 0.0f;
    if ((unsigned)gh < HH && (unsigned)gw < WW)
      v = x[(((size_t)b * CF + ci) * HH + gh) * WW + gw];
    xs[row][ci][wl] = v;
  }
  __syncthreads();

  const int m = wave * 16 + lrow;  // A-matrix row (output channel)
  v8f accQ0 = {}, accQ1 = {}, accK0 = {}, accK1 = {}, accV0 = {}, accV1 = {};

  // q and mkey: 1x1 conv (center row of the tile, kx offset 1)
  for (int k0 = 0; k0 < CF; k0 += 4) {
    int k = k0 + khalf;
    v2f aq = *(const v2f*)(wf + (size_t)m * CF + k);
    v2f ak = *(const v2f*)(wk + (size_t)m * CF + k);
    v2f b0v = { xs[1][k][1 + lrow],  xs[1][k + 1][1 + lrow]  };
    v2f b1v = { xs[1][k][17 + lrow], xs[1][k + 1][17 + lrow] };
    accQ0 = wmma4(aq, b0v, accQ0);
    accQ1 = wmma4(aq, b1v, accQ1);
    accK0 = wmma4(ak, b0v, accK0);
    accK1 = wmma4(ak, b1v, accK1);
  }
  // vals: 3x3 conv as 9 accumulating tap-GEMMs
  for (int tap = 0; tap < 9; ++tap) {
    int ky = tap / 3, kx = tap % 3;
    const float* wrow = wvT + ((size_t)tap * CC + m) * CF;
    for (int k0 = 0; k0 < CF; k0 += 4) {
      int k = k0 + khalf;
      v2f av = *(const v2f*)(wrow + k);
      v2f b0v = { xs[ky][k][lrow + kx],      xs[ky][k + 1][lrow + kx]      };
      v2f b1v = { xs[ky][k][16 + lrow + kx], xs[ky][k + 1][16 + lrow + kx] };
      accV0 = wmma4(av, b0v, accV0);
      accV1 = wmma4(av, b1v, accV1);
    }
  }

  // epilogue: add bias, store q/vals, reduce mkey.q into lmain via LDS atomics
  const int mhalf = (lane >> 4) * 8;
  float p0 = 0.0f, p1 = 0.0f;
  for (int r = 0; r < 8; ++r) {
    int mo = wave * 16 + r + mhalf;
    float bq = bf[mo], bkk = bk[mo], bvv = bv[mo];
    float q0 = accQ0[r] + bq, q1 = accQ1[r] + bq;
    float k0v = accK0[r] + bkk, k1v = accK1[r] + bkk;
    float v0 = accV0[r] + bvv, v1 = accV1[r] + bvv;
    size_t base = (((size_t)b * CC + mo) * HH + h) * WW + w0;
    q_out[base + lrow]      = q0;
    q_out[base + 16 + lrow] = q1;
    v_out[base + lrow]      = v0;
    v_out[base + 16 + lrow] = v1;
    p0 += k0v * q0;
    p1 += k1v * q1;
  }
  atomicAdd(&lm_s[lrow], p0);
  atomicAdd(&lm_s[16 + lrow], p1);
  __syncthreads();
  if (tid < 32)
    lmain_out[((size_t)b * HH + h) * WW + w0 + tid] = lm_s[tid];
}

// ---------------------------------------------------------------------------
// Kernel B: fused ckeys-GEMM + cross-attention + weighted combine.
// One WG = 16 coarse pixels = 64 subpixels (2 rows x 32 cols) for one (b,h).
// Context tiles (3 x 128 x 64 f32) arrive via global_load_async_to_lds_b128,
// feed the Wc GEMM (M=128,N=64,K=128) AND the final raw-feature combine, so
// contexts are read from HBM exactly once.
// ---------------------------------------------------------------------------
__global__ void __launch_bounds__(256)
attn_kernel(const float* __restrict__ ctx,
            const float* __restrict__ wc, const float* __restrict__ bc,
            const float* __restrict__ q_ws, const float* __restrict__ v_ws,
            const float* __restrict__ lmain_ws,
            float* __restrict__ out) {
  const int w0  = blockIdx.x * 16;   // coarse col base
  const int sc0 = w0 * 2;            // subpixel col base
  const int h = blockIdx.y;
  const int b = blockIdx.z;
  const int tid  = threadIdx.x;
  const int wave = tid >> 5, lane = tid & 31;
  const int lrow = lane & 15;
  const int khalf = (lane >> 4) * 2;

  __shared__ float cs[NCTX][CC][72];    // padded to kill bank conflicts
  __shared__ float logit_s[NCTX][64];
  __shared__ float attn_s[4][64];

  if (tid < NCTX * 64) ((float*)logit_s)[tid] = 0.0f;

  // ---- async tensor staging: 16B per lane per issue, ASYNCcnt tracked ----
  unsigned cs_base = (unsigned)(uintptr_t)(&cs[0][0][0]);
  for (int idx4 = tid; idx4 < (NCTX * CC * 64) / 4; idx4 += 256) {
    int f0  = idx4 * 4;
    int n   = f0 / (CC * 64);
    int rem = f0 % (CC * 64);
    int c   = rem / 64;
    int sp  = rem % 64;            // sp = i*32 + j, j%4 == 0 here
    int i = sp >> 5, j = sp & 31;
    const float* gp = ctx +
        ((((size_t)n * BB + b) * CC + c) * (2 * HH) + (2 * h + i)) * (2 * WW) + sc0 + j;
    unsigned lds = cs_base + (unsigned)(((n * CC + c) * 72 + sp) * 4);
    asm volatile("global_load_async_to_lds_b128 %0, %1, off"
                 :: "v"(lds), "v"(gp) : "memory");
  }
  asm volatile("s_wait_asynccnt 0x0" ::: "memory");
  __syncthreads();

  // per-lane q values q[m(r), coarse_pixel(col)] reused across all 3 contexts
  const int mhalf = (lane >> 4) * 8;
  float qreg[4][8];
  for (int t = 0; t < 4; ++t) {
    int sp = t * 16 + lrow;
    int j = sp & 31;
    int wcoarse = w0 + (j >> 1);
    for (int r = 0; r < 8; ++r) {
      int mo = wave * 16 + r + mhalf;
      qreg[t][r] = q_ws[(((size_t)b * CC + mo) * HH + h) * WW + wcoarse];
    }
  }

  const float* wrow = wc + (size_t)(wave * 16 + lrow) * CC;
  for (int n = 0; n < NCTX; ++n) {
    v8f acc0 = {}, acc1 = {}, acc2 = {}, acc3 = {};
    for (int k0 = 0; k0 < CC; k0 += 4) {
      int k = k0 + khalf;
      v2f a = *(const v2f*)(wrow + k);
      v2f b0v = { cs[n][k][lrow],      cs[n][k + 1][lrow]      };
      v2f b1v = { cs[n][k][16 + lrow], cs[n][k + 1][16 + lrow] };
      v2f b2v = { cs[n][k][32 + lrow], cs[n][k + 1][32 + lrow] };
      v2f b3v = { cs[n][k][48 + lrow], cs[n][k + 1][48 + lrow] };
      acc0 = wmma4(a, b0v, acc0);
      acc1 = wmma4(a, b1v, acc1);
      acc2 = wmma4(a, b2v, acc2);
      acc3 = wmma4(a, b3v, acc3);
    }
    // logits: (Wc*ctx + bc) . q, reduced over all 128 channels via LDS atomics
    float p0 = 0, p1 = 0, p2 = 0, p3 = 0;
    for (int r = 0; r < 8; ++r) {
      int mo = wave * 16 + r + mhalf;
      float bias = bc[mo];
      p0 += (acc0[r] + bias) * qreg[0][r];
      p1 += (acc1[r] + bias) * qreg[1][r];
      p2 += (acc2[r] + bias) * qreg[2][r];
      p3 += (acc3[r] + bias) * qreg[3][r];
    }
    atomicAdd(&logit_s[n][lrow],      p0);
    atomicAdd(&logit_s[n][16 + lrow], p1);
    atomicAdd(&logit_s[n][32 + lrow], p2);
    atomicAdd(&logit_s[n][48 + lrow], p3);
  }
  __syncthreads();

  // 4-way softmax per subpixel (3 contexts + mainline)
  if (tid < 64) {
    int sp = tid;
    int j = sp & 31;
    float l0 = logit_s[0][sp], l1 = logit_s[1][sp], l2 = logit_s[2][sp];
    float lm = lmain_ws[((size_t)b * HH + h) * WW + w0 + (j >> 1)];
    float mx = fmaxf(fmaxf(l0, l1), fmaxf(l2, lm));
    float e0 = __expf(l0 - mx), e1 = __expf(l1 - mx);
    float e2 = __expf(l2 - mx), em = __expf(lm - mx);
    float inv = 1.0f / (e0 + e1 + e2 + em);
    attn_s[0][sp] = e0 * inv;
    attn_s[1][sp] = e1 * inv;
    attn_s[2][sp] = e2 * inv;
    attn_s[3][sp] = em * inv;
  }
  __syncthreads();

  // weighted combine: raw context features (still in LDS) + mainline values
  for (int idx = tid; idx < CC * 64; idx += 256) {
    int c  = idx >> 6;
    int sp = idx & 63;
    int i = sp >> 5, j = sp & 31;
    float vmain = v_ws[(((size_t)b * CC + c) * HH + h) * WW + w0 + (j >> 1)];
    float o = attn_s[0][sp] * cs[0][c][sp]
            + attn_s[1][sp] * cs[1][c][sp]
            + attn_s[2][sp] * cs[2][c][sp]
            + attn_s[3][sp] * vmain;
    out[(((size_t)b * CC + c) * (2 * HH) + (2 * h + i)) * (2 * WW) + sc0 + j] = o;
  }
}

// ---------------------------------------------------------------------------
extern "C" void kernel_launch(void* const* d_in, const int* in_sizes, int n_in,
                              void* d_out, int out_size, void* d_ws, size_t ws_size,
                              hipStream_t stream) {
  const float* contexts   = (const float*)d_in[0];
  const float* mainstream = (const float*)d_in[1];
  const float* Wc_w = (const float*)d_in[2];
  const float* Wc_b = (const float*)d_in[3];
  const float* Wf_w = (const float*)d_in[4];
  const float* Wf_b = (const float*)d_in[5];
  const float* Wk_w = (const float*)d_in[6];
  const float* Wk_b = (const float*)d_in[7];
  const float* Wv_w = (const float*)d_in[8];
  const float* Wv_b = (const float*)d_in[9];
  float* out = (float*)d_out;

  float* ws    = (float*)d_ws;
  float* q_ws  = ws;                                   // B*CC*H*W
  float* v_ws  = q_ws + (size_t)BB * CC * HH * WW;     // B*CC*H*W
  float* lm_ws = v_ws + (size_t)BB * CC * HH * WW;     // B*H*W
  float* wvT   = lm_ws + (size_t)BB * HH * WW;         // 9*CC*CF

  hipLaunchKernelGGL(wv_transpose_kernel,
                     dim3((CC * CF * 9 + 255) / 256), dim3(256), 0, stream,
                     Wv_w, wvT);
  hipLaunchKernelGGL(proj_kernel,
                     dim3(WW / 32, HH, BB), dim3(256), 0, stream,
                     mainstream, Wf_w, Wf_b, Wk_w, Wk_b, wvT, Wv_b,
                     q_ws, v_ws, lm_ws);
  hipLaunchKernelGGL(attn_kernel,
                     dim3(WW / 16, HH, BB), dim3(256), 0, stream,
                     contexts, Wc_w, Wc_b, q_ws, v_ws, lm_ws, out);
}